// TopKLayer_27874337751523
// MI455X (gfx1250) — compile-verified
//
#include <hip/hip_runtime.h>
#include <hip/hip_bf16.h>
#include <stdint.h>

#define HW     3136          // 56*56
#define HW4    (HW / 4)      // 784 float4 chunks
#define KTOP   627           // floor(0.2 * 3136)
#define NROWS  (32 * 256)    // n * c
#define BLOCK  256           // 8 wave32 waves

#if defined(__HIP_DEVICE_COMPILE__)
typedef int v4i __attribute__((ext_vector_type(4)));
typedef __attribute__((address_space(1))) v4i g_v4i;   // global int4
typedef __attribute__((address_space(3))) v4i l_v4i;   // LDS int4
#endif

__global__ __launch_bounds__(BLOCK)
void topk_mask_kernel(const float* __restrict__ x, float* __restrict__ out) {
    __shared__ float    sdata[HW];
    __shared__ unsigned s_hist[256];
    __shared__ unsigned s_kth;     // remaining rank within current prefix class
    __shared__ unsigned s_thresh;  // accumulated |x| bit-pattern of k-th largest
    __shared__ unsigned s_tie;     // tie-break counter at threshold

    const int    t    = threadIdx.x;
    const int    row  = blockIdx.x;
    const size_t base = (size_t)row * HW;

    // ---------------- global row -> LDS ----------------
    // CDNA5: GLOBAL_LOAD_ASYNC_TO_LDS_B128 (ASYNCcnt), no VGPR staging.
    const float4* __restrict__ src4 = (const float4*)(x + base);
#if defined(__HIP_DEVICE_COMPILE__) && __has_builtin(__builtin_amdgcn_global_load_async_to_lds_b128)
    for (int i = t; i < HW4; i += BLOCK) {
        __builtin_amdgcn_global_load_async_to_lds_b128(
            (g_v4i*)(src4 + i),
            (l_v4i*)&sdata[i * 4],
            /*offset=*/0, /*cpol=*/0);
    }
  #if __has_builtin(__builtin_amdgcn_s_wait_asynccnt)
    __builtin_amdgcn_s_wait_asynccnt(0);
  #else
    asm volatile("s_wait_asynccnt 0" ::: "memory");
  #endif
#else
    for (int i = t; i < HW4; i += BLOCK) {
        float4 v = src4[i];
        *(float4*)&sdata[i * 4] = v;
    }
#endif
    if (t == 0) { s_kth = KTOP; s_thresh = 0u; s_tie = 0u; }
    __syncthreads();

    // ---------------- 4-pass radix select on bits of |x| ----------------
    // |x| bit patterns (x & 0x7fffffff) order identically to values for
    // non-negative floats: byte-wise MSB-first select gives the exact
    // k-th largest magnitude.
    #pragma unroll
    for (int pass = 0; pass < 4; ++pass) {
        const int      shift    = 24 - pass * 8;
        const unsigned prefmask = (pass == 0) ? 0u : (0xFFFFFFFFu << (shift + 8));
        const unsigned pref     = s_thresh;   // higher bytes chosen so far (stable: barrier-ordered)

        s_hist[t] = 0u;
        __syncthreads();

        // Vectorized LDS scan: ds_load_b128, 4 elements per access.
        for (int i = t; i < HW4; i += BLOCK) {
            const float4 v = *(const float4*)&sdata[i * 4];
            #pragma unroll
            for (int c = 0; c < 4; ++c) {
                const float    f = (c == 0) ? v.x : (c == 1) ? v.y : (c == 2) ? v.z : v.w;
                const unsigned u = __float_as_uint(f) & 0x7FFFFFFFu;
                if ((u & prefmask) == pref)
                    atomicAdd(&s_hist[(u >> shift) & 0xFFu], 1u);
            }
        }
        __syncthreads();

        // Single-wave32 bin selection: lane l owns bins [8l, 8l+7].
        if (t < 32) {
            const unsigned kth = s_kth;
            unsigned h[8];
            unsigned localSum = 0u;
            #pragma unroll
            for (int j = 0; j < 8; ++j) { h[j] = s_hist[t * 8 + j]; localSum += h[j]; }

            // Inclusive suffix-sum over lanes: laneSuffix(l) = sum_{l' >= l} localSum(l')
            unsigned laneSuffix = localSum;
            #pragma unroll
            for (int off = 1; off < 32; off <<= 1) {
                unsigned w = __shfl_down(laneSuffix, off, 32);
                if (t + off < 32) laneSuffix += w;
            }
            const unsigned hi = laneSuffix - localSum;  // count in bins above this lane's range

            // Walk own 8 bins from high to low; exactly one (lane,bin) wins.
            unsigned run = hi;
            #pragma unroll
            for (int j = 7; j >= 0; --j) {
                const unsigned above = run;       // count strictly above bin (8t+j)
                run += h[j];                      // suffix including this bin
                if (run >= kth && above < kth) {
                    s_thresh = pref | ((unsigned)(t * 8 + j) << shift);
                    s_kth    = kth - above;       // rank within the equal-byte class
                }
            }
        }
        __syncthreads();
    }

    // ---------------- masked write-out (b128 stores) ----------------
    const unsigned T     = s_thresh;  // exact bits of k-th largest |x|
    const unsigned keepT = s_kth;     // how many == T survive (ties)
    float4* __restrict__ dst4 = (float4*)(out + base);

    for (int i = t; i < HW4; i += BLOCK) {
        float4 v = *(const float4*)&sdata[i * 4];
        float4 o;
        float*       ov = (float*)&o;
        const float* iv = (const float*)&v;
        #pragma unroll
        for (int c = 0; c < 4; ++c) {
            const float    f = iv[c];
            const unsigned u = __float_as_uint(f) & 0x7FFFFFFFu;
            float          r = 0.0f;
            if (u > T) {
                r = f;
            } else if (u == T) {
                unsigned pos = atomicAdd(&s_tie, 1u);
                if (pos < keepT) r = f;
            }
            ov[c] = r;
        }
        dst4[i] = o;
    }
}

extern "C" void kernel_launch(void* const* d_in, const int* in_sizes, int n_in,
                              void* d_out, int out_size, void* d_ws, size_t ws_size,
                              hipStream_t stream) {
    const float* x   = (const float*)d_in[0];
    float*       out = (float*)d_out;
    (void)in_sizes; (void)n_in; (void)out_size; (void)d_ws; (void)ws_size;

    dim3 grid(NROWS);
    dim3 block(BLOCK);
    topk_mask_kernel<<<grid, block, 0, stream>>>(x, out);
}